// BehaviorCloner_43748536877193
// MI455X (gfx1250) — compile-verified
//
#include <hip/hip_runtime.h>
#include <stdint.h>

#define NEL 14
#define NCH 20
#define CAC 17
#define CLN 13
#define NB  32
#define HH  256
#define WW  256
#define HWP (HH*WW)
#define TILE 32
#define SR 40          // state region (halo 4)
#define SP 41          // padded row
#define LR 42          // label region (halo 5)
#define LP 43
#define NTHREADS 256

#if defined(__has_builtin)
#if __has_builtin(__builtin_amdgcn_global_load_async_to_lds_b32)
#define USE_ASYNC_LDS 1
#endif
#endif

typedef __attribute__((address_space(1))) int* gas1_t;
typedef __attribute__((address_space(3))) int* las3_t;

__global__ __launch_bounds__(NTHREADS)
void bc_fused_kernel(const float* __restrict__ w, const float* __restrict__ evg,
                     const float* __restrict__ info, float* __restrict__ out)
{
    __shared__ float ev[NEL * NCH];           // elem_vecs copy
    __shared__ float caInit[SR * SP];         // staged ch17 patch (async target)
    __shared__ float s0a[SR * SP], s13a[SR * SP], s17a[SR * SP];
    __shared__ float s0b[SR * SP], s13b[SR * SP], s17b[SR * SP];
    __shared__ unsigned char lab[LR * LP];    // argmax labels (0..13)
    __shared__ unsigned char caf[SR * SP];    // ca_final (0..13)
    __shared__ signed char osa[SR * SP], osb[SR * SP]; // last overwrite src, -1 = none

    const int tid = threadIdx.x;
    const int b   = blockIdx.z;
    const int ty0 = blockIdx.y * TILE;
    const int tx0 = blockIdx.x * TILE;
    const int bbase = b * NCH * HWP;

    // ---- Phase 0: async-stage ca_init (ch17) halo-4 patch into LDS ----
    for (int i = tid; i < SR * SR; i += NTHREADS) {
        int r = i / SR, c = i % SR;
        int gy = (ty0 + r - 4) & (HH - 1);
        int gx = (tx0 + c - 4) & (WW - 1);
        int gidx = bbase + CAC * HWP + gy * WW + gx;
#ifdef USE_ASYNC_LDS
        __builtin_amdgcn_global_load_async_to_lds_b32(
            (gas1_t)(uintptr_t)(w + gidx),
            (las3_t)(uint32_t)(uintptr_t)&caInit[r * SP + c],
            0, 0);
#else
        caInit[r * SP + c] = w[gidx];
#endif
    }

    for (int i = tid; i < NEL * NCH; i += NTHREADS) ev[i] = evg[i];

    // ---- Phase 1: labels (argmax ch0..13) over halo-5 region; stash s0,s13 ----
    for (int i = tid; i < LR * LR; i += NTHREADS) {
        int r = i / LR, c = i % LR;
        int gy = (ty0 + r - 5) & (HH - 1);
        int gx = (tx0 + c - 5) & (WW - 1);
        int p = bbase + gy * WW + gx;
        float v0 = w[p];
        float best = v0; int bi = 0;
        float v13 = 0.f;
#pragma unroll
        for (int ch = 1; ch < NEL; ++ch) {
            float v = w[p + ch * HWP];
            if (ch == CLN) v13 = v;
            if (v > best) { best = v; bi = ch; }   // first-max, like jnp.argmax
        }
        lab[r * LP + c] = (unsigned char)bi;
        if (r >= 1 && r <= SR && c >= 1 && c <= SR) {
            int sq = (r - 1) * SP + (c - 1);
            s0a[sq]  = v0;
            s13a[sq] = v13;
        }
    }

#ifdef USE_ASYNC_LDS
#if __has_builtin(__builtin_amdgcn_s_wait_asynccnt)
    __builtin_amdgcn_s_wait_asynccnt(0);
#else
    asm volatile("s_wait_asynccnt 0" ::: "memory");
#endif
#endif
    __syncthreads();

    // ---- Phase 2: ca_final (sequential 4-neighbor pull), init s17 / osrc ----
    for (int i = tid; i < SR * SR; i += NTHREADS) {
        int r = i / SR, c = i % SR;
        int lr = r + 1, lc = c + 1;
        int sq = r * SP + c;
        float ca = caInit[sq];
        if (!(s13a[sq] > 0.5f)) {
            ca = 0.f;                               // non-cloner -> 0
        } else {
            if (ca == 0.f || ca == 13.f) ca = (float)lab[(lr + 1) * LP + lc]; // below
            if (ca == 0.f || ca == 13.f) ca = (float)lab[(lr - 1) * LP + lc]; // above
            if (ca == 0.f || ca == 13.f) ca = (float)lab[lr * LP + lc - 1];   // left
            if (ca == 0.f || ca == 13.f) ca = (float)lab[lr * LP + lc + 1];   // right
        }
        caf[sq]  = (unsigned char)(int)ca;
        s17a[sq] = ca;
        osa[sq]  = -1;
    }
    __syncthreads();

    // ---- Phase 3: 4 simultaneous-update propagation steps (ping-pong) ----
    float *A0 = s0a, *A13 = s13a, *A17 = s17a;
    float *B0 = s0b, *B13 = s13b, *B17 = s17b;
    signed char *OA = osa, *OB = osb;
    const int drs[4] = { 1, -1, 0, 0 };
    const int dcs[4] = { 0, 0, -1, 1 };
#pragma unroll
    for (int k = 0; k < 4; ++k) {
        const int lo = k + 1, hi = SR - 2 - k;
        const int dr = drs[k], dc = dcs[k];
        for (int i = tid; i < SR * SR; i += NTHREADS) {
            int r = i / SR, c = i % SR;
            int p = r * SP + c;
            float c0 = A0[p], c13 = A13[p], c17 = A17[p];
            signed char os = OA[p];
            if (r >= lo && r <= hi && c >= lo && c <= hi) {
                int q = (r + dr) * SP + (c + dc);
                float n13 = A13[q], n17 = A17[q];
                bool act = (n13 > 0.5f) && (n17 != 0.f) && (n17 != 13.f);
                if (act && (c0 > 0.5f)) {
                    int e = caf[q];
                    c0  = ev[e * NCH + 0];
                    c13 = ev[e * NCH + CLN];
                    c17 = ev[e * NCH + CAC];
                    os  = (signed char)e;
                }
            }
            B0[p] = c0; B13[p] = c13; B17[p] = c17; OB[p] = os;
        }
        __syncthreads();
        float* t;
        t = A0;  A0 = B0;  B0 = t;
        t = A13; A13 = B13; B13 = t;
        t = A17; A17 = B17; B17 = t;
        signed char* u = OA; OA = OB; OB = u;
    }

    // ---- Phase 4: write all 20 output channels for the interior tile ----
    for (int i = tid; i < TILE * TILE; i += NTHREADS) {
        int r = i / TILE, c = i % TILE;
        int sq = (r + 4) * SP + (c + 4);
        int gy = ty0 + r, gx = tx0 + c;
        int p = bbase + gy * WW + gx;
        int os = (int)OA[sq];
        float cafv = (float)caf[sq];
#pragma unroll
        for (int ch = 0; ch < NCH; ++ch) {
            float v = w[p + ch * HWP];
            if (ch == CAC) v = cafv;
            if (os >= 0) v = ev[os * NCH + ch];
            out[p + ch * HWP] = v;
        }
    }

    // ---- info passthrough (tuple output: world then info) ----
    if (blockIdx.x == 0 && blockIdx.y == 0 && tid == 0) {
        out[NB * NCH * HWP + b] = info[b];
    }
}

extern "C" void kernel_launch(void* const* d_in, const int* in_sizes, int n_in,
                              void* d_out, int out_size, void* d_ws, size_t ws_size,
                              hipStream_t stream) {
    (void)in_sizes; (void)n_in; (void)out_size; (void)d_ws; (void)ws_size;
    const float* w    = (const float*)d_in[0];
    const float* evg  = (const float*)d_in[1];
    const float* info = (const float*)d_in[2];
    float* out = (float*)d_out;
    dim3 grid(WW / TILE, HH / TILE, NB);
    bc_fused_kernel<<<grid, NTHREADS, 0, stream>>>(w, evg, info, out);
}